// CapsuleLayer_51608327029174
// MI455X (gfx1250) — compile-verified
//
#include <hip/hip_runtime.h>
#include <hip/hip_bf16.h>
#include <math.h>

// Problem dims (fixed by the reference)
#define BB      64      // batch
#define NR      2304    // route nodes
#define CC      32      // capsules
#define OO      16      // out dim
#define II      8       // in dim
#define NCHUNK  128     // n-tile staged in LDS for the s-kernel
#define RS      (NCHUNK * II + 4)   // padded LDS row stride (floats), conflict-free

typedef __attribute__((ext_vector_type(2))) float v2f;
typedef __attribute__((ext_vector_type(8))) float v8f;

// ---------------------------------------------------------------------------
// Kernel 1: softmax over n (dim=1) per capsule c.  c_ij[n,c] = softmax_n(b_ij[n,c])
// One block per c. b_ij/c_ij layout: [N, C] row-major.
// ---------------------------------------------------------------------------
__global__ __launch_bounds__(256)
void caps_softmax_n(const float* __restrict__ bij, float* __restrict__ cij) {
    const int c   = blockIdx.x;
    const int tid = threadIdx.x;
    __shared__ float red[256];

    float m = -3.4e38f;
    for (int n = tid; n < NR; n += 256) m = fmaxf(m, bij[(size_t)n * CC + c]);
    red[tid] = m;
    __syncthreads();
    for (int s = 128; s > 0; s >>= 1) {
        if (tid < s) red[tid] = fmaxf(red[tid], red[tid + s]);
        __syncthreads();
    }
    m = red[0];
    __syncthreads();

    float sum = 0.0f;
    for (int n = tid; n < NR; n += 256) sum += __expf(bij[(size_t)n * CC + c] - m);
    red[tid] = sum;
    __syncthreads();
    for (int s = 128; s > 0; s >>= 1) {
        if (tid < s) red[tid] += red[tid + s];
        __syncthreads();
    }
    const float inv = 1.0f / red[0];

    for (int n = tid; n < NR; n += 256)
        cij[(size_t)n * CC + c] = __expf(bij[(size_t)n * CC + c] - m) * inv;
}

// ---------------------------------------------------------------------------
// Kernel 2: s_j + squash, with u_hat recomputed on the fly via f32 WMMA.
//   s[b,c,o] = sum_n c_ij[n,c] * sum_i W[n,c,o,i] * x[b,n,i]
//   v[b,c,o] = s^3 / ((1+s^2)*sqrt(s^2))
// Grid: (C, B/16). Block: 128 threads = 4 waves; wave w covers n = w*32..w*32+31
// of each 128-n chunk. Each wave holds one 16(b)x16(o) WMMA accumulator tile;
// partial tiles are reduced through LDS at the end.
// ---------------------------------------------------------------------------
__global__ __launch_bounds__(128)
void caps_s_squash(const float* __restrict__ x, const float* __restrict__ W,
                   const float* __restrict__ cij, float* __restrict__ vout) {
    __shared__ float xs[16 * RS];   // 65,792 B of the 320 KB WGP LDS

    const int c    = blockIdx.x;
    const int b0   = blockIdx.y * 16;
    const int tid  = threadIdx.x;
    const int wid  = tid >> 5;
    const int lane = tid & 31;
    const int bl   = lane & 15;   // WMMA M index (batch row)
    const int hi   = lane >> 4;   // half-wave select

    v8f acc = {};

    for (int nc0 = 0; nc0 < NR; nc0 += NCHUNK) {
        __syncthreads();
        // Stage x[b0..b0+15, nc0..nc0+127, 0..7] into LDS (coalesced float4).
        for (int idx = tid; idx < 16 * (NCHUNK * II / 4); idx += 128) {
            const int row  = idx >> 8;    // /256
            const int col4 = idx & 255;
            const float4 val = *(const float4*)(x + (((size_t)(b0 + row)) * NR + nc0) * II + col4 * 4);
            *(float4*)(xs + row * RS + col4 * 4) = val;
        }
        __syncthreads();

        for (int j = 0; j < 32; ++j) {
            const int nl = wid * 32 + j;
            const int n  = nc0 + nl;
            const float cv = cij[(size_t)n * CC + c];   // uniform across wave

            // A fragments: x[b, n, k] * c_ij[n,c]; 16x4 f32 layout:
            //   v0 = k{0|2}, v1 = k{1|3} (lanes 0-15 | 16-31)
            const float* xr = xs + bl * RS + nl * II + hi * 2;
            const float2 a0 = *(const float2*)(xr);        // k = hi*2, hi*2+1
            const float2 a1 = *(const float2*)(xr + 4);    // k = 4+hi*2, 5+hi*2
            v2f A0 = { a0.x * cv, a0.y * cv };
            v2f A1 = { a1.x * cv, a1.y * cv };

            // B fragments: W[n,c,o,k]; 4x16 layout: v0 row k=hi, v1 row k=2+hi
            const float* wp = W + (((size_t)n * CC + c) * OO + bl) * II;
            const float4 q0 = *(const float4*)(wp);        // k 0..3
            const float4 q1 = *(const float4*)(wp + 4);    // k 4..7
            v2f B0 = { hi ? q0.y : q0.x, hi ? q0.w : q0.z };
            v2f B1 = { hi ? q1.y : q1.x, hi ? q1.w : q1.z };

            acc = __builtin_amdgcn_wmma_f32_16x16x4_f32(false, A0, false, B0,
                                                        (short)0, acc, false, false);
            acc = __builtin_amdgcn_wmma_f32_16x16x4_f32(false, A1, false, B1,
                                                        (short)0, acc, false, false);
        }
    }

    // Reduce the 4 per-wave partial tiles through LDS, squash, store.
    __syncthreads();
    float* red = xs;
#pragma unroll
    for (int r = 0; r < 8; ++r)
        red[(wid * 32 + lane) * 8 + r] = acc[r];
    __syncthreads();

    for (int f = tid; f < 256; f += 128) {
        const float s  = red[f] + red[256 + f] + red[512 + f] + red[768 + f];
        const float sq = s * s;
        const float v  = (sq * s) / ((1.0f + sq) * sqrtf(sq));   // literal squash
        const int ln = f >> 3;
        const int r  = f & 7;
        const int b  = b0 + r + 8 * (ln >> 4);
        const int o  = ln & 15;
        vout[((size_t)b * CC + c) * OO + o] = v;   // == output layout [B,C,O]
    }
}

// ---------------------------------------------------------------------------
// Kernel 3: b_ij[n,c] += (1/B) * sum_b sum_o u_hat[b,n,c,o] * v[b,c,o]
// One thread per (n,c); W[n,c,:,:] (128 f32) held in registers, u_hat
// recomputed per b. W read exactly once (37.7 MB), x/v served from cache.
// ---------------------------------------------------------------------------
__global__ __launch_bounds__(256)
void caps_bupdate(const float* __restrict__ x, const float* __restrict__ W,
                  const float* __restrict__ v, float* __restrict__ bij) {
    const int idx = blockIdx.x * blockDim.x + threadIdx.x;   // n*32 + c
    if (idx >= NR * CC) return;
    const int n = idx >> 5;
    const int c = idx & 31;

    const float* wp = W + (size_t)idx * (OO * II);
    float w[OO * II];
#pragma unroll
    for (int q = 0; q < (OO * II) / 4; ++q) {
        const float4 t = *(const float4*)(wp + q * 4);
        w[q * 4 + 0] = t.x; w[q * 4 + 1] = t.y;
        w[q * 4 + 2] = t.z; w[q * 4 + 3] = t.w;
    }

    float acc = 0.0f;
    for (int b = 0; b < BB; ++b) {
        const float* xp = x + ((size_t)b * NR + n) * II;
        const float4 x0 = *(const float4*)(xp);
        const float4 x1 = *(const float4*)(xp + 4);
        const float* vp = v + ((size_t)b * CC + c) * OO;
        float a = 0.0f;
#pragma unroll
        for (int o = 0; o < OO; ++o) {
            const float* wr = w + o * II;
            float u = wr[0] * x0.x + wr[1] * x0.y + wr[2] * x0.z + wr[3] * x0.w
                    + wr[4] * x1.x + wr[5] * x1.y + wr[6] * x1.z + wr[7] * x1.w;
            a += u * vp[o];
        }
        acc += a;
    }
    bij[idx] += acc * (1.0f / (float)BB);
}

// ---------------------------------------------------------------------------
// Driver: 3 routing iterations; v from iterations 0/1 lives in ws, final v
// goes straight to d_out (identical [B,C,O] layout as the reference output).
// ---------------------------------------------------------------------------
extern "C" void kernel_launch(void* const* d_in, const int* in_sizes, int n_in,
                              void* d_out, int out_size, void* d_ws, size_t ws_size,
                              hipStream_t stream) {
    const float* x = (const float*)d_in[0];   // [B, N, I]
    const float* W = (const float*)d_in[1];   // [N, C, O, I]
    float* out = (float*)d_out;               // [B, C, O, 1]

    float* bij  = (float*)d_ws;               // [N, C]
    float* cij  = bij + (size_t)NR * CC;      // [N, C]
    float* vbuf = cij + (size_t)NR * CC;      // [B, C, O]

    hipMemsetAsync(bij, 0, (size_t)NR * CC * sizeof(float), stream);

    for (int it = 0; it < 3; ++it) {
        caps_softmax_n<<<CC, 256, 0, stream>>>(bij, cij);
        float* vdst = (it == 2) ? out : vbuf;
        caps_s_squash<<<dim3(CC, BB / 16), 128, 0, stream>>>(x, W, cij, vdst);
        if (it < 2) {
            caps_bupdate<<<(NR * CC) / 256, 256, 0, stream>>>(x, W, vbuf, bij);
        }
    }
}